// CommonModel_37933151158782
// MI455X (gfx1250) — compile-verified
//
#include <hip/hip_runtime.h>

#define Hdim   64
#define NG_    50000
#define NS_    50000
#define E_TOT  1000000

typedef __attribute__((ext_vector_type(2))) float v2f;
typedef __attribute__((ext_vector_type(8))) float v8f;

// ---------------------------------------------------------------------------
// monotone float <-> uint encoding for atomic segment-max
// ---------------------------------------------------------------------------
__device__ __forceinline__ unsigned enc_f(float f) {
    unsigned b = __float_as_uint(f);
    return (b & 0x80000000u) ? ~b : (b | 0x80000000u);
}
__device__ __forceinline__ float dec_f(unsigned u) {
    unsigned b = (u & 0x80000000u) ? (u & 0x7FFFFFFFu) : ~u;
    return __uint_as_float(b);
}

// ---------------------------------------------------------------------------
// Fused SAGE / plain GEMM via V_WMMA_F32_16X16X4_F32.
// out[N, 64] = relu?( scale(A1) @ WL^T  (+ A2 @ WR^T) (+ bias) )
// scale(A1) row i multiplied by 1/max(cnt[i],1) when cnt != nullptr.
// Block = 128 threads (4 waves); wave w computes output columns [16w,16w+16);
// grid.x = ceil(N/16) row tiles.  In-place out==A1/A2 is safe (syncthreads
// before store; each row tile is touched only by its own block).
// ---------------------------------------------------------------------------
__global__ __launch_bounds__(128)
void sage_wmma(const float* __restrict__ A1, const int* __restrict__ cnt,
               const float* __restrict__ WL,
               const float* __restrict__ A2, const float* __restrict__ WR,
               const float* __restrict__ bias, float* __restrict__ out,
               int N, int K, int do_relu)
{
    const int lane = threadIdx.x & 31;
    const int wave = threadIdx.x >> 5;
    const int half = lane >> 4;      // 0: lanes 0-15, 1: lanes 16-31
    const int m16  = lane & 15;
    const int colBase = wave * 16;
    const int ncol = colBase + m16;  // B column / output column (N dim of tile)
    const int row  = blockIdx.x * 16 + m16;
    const int rc   = row < N ? row : N - 1;   // clamped load row

    v8f c = {};
    float s1 = 1.0f;
    if (cnt) s1 = 1.0f / fmaxf((float)cnt[rc], 1.0f);

    const float* a1p = A1 + (size_t)rc * K;
    for (int k0 = 0; k0 < K; k0 += 4) {
        const int ka = k0 + half * 2;          // A: K = vgpr + 2*laneHalf
        v2f a, b;
        a.x = a1p[ka]     * s1;
        a.y = a1p[ka + 1] * s1;
        b.x = WL[(size_t)ncol * K + ka];       // B[k][n] = W[n*K + k]
        b.y = WL[(size_t)ncol * K + ka + 1];
        c = __builtin_amdgcn_wmma_f32_16x16x4_f32(false, a, false, b,
                                                  (short)0, c, false, false);
    }
    if (A2) {
        const float* a2p = A2 + (size_t)rc * K;
        for (int k0 = 0; k0 < K; k0 += 4) {
            const int ka = k0 + half * 2;
            v2f a, b;
            a.x = a2p[ka];
            a.y = a2p[ka + 1];
            b.x = WR[(size_t)ncol * K + ka];
            b.y = WR[(size_t)ncol * K + ka + 1];
            c = __builtin_amdgcn_wmma_f32_16x16x4_f32(false, a, false, b,
                                                      (short)0, c, false, false);
        }
    }

    __syncthreads();   // all waves finished reading A rows -> in-place safe

    const float bv = bias ? bias[ncol] : 0.0f;
#pragma unroll
    for (int r = 0; r < 8; ++r) {
        // C/D layout: vgpr r, lanes0-15 -> M=r ; lanes16-31 -> M=r+8
        const int orow = blockIdx.x * 16 + r + half * 8;
        if (orow < N) {
            float v = c[r] + bv;
            if (do_relu) v = fmaxf(v, 0.0f);
            out[(size_t)orow * Hdim + ncol] = v;
        }
    }
}

// ---------------------------------------------------------------------------
// scatter-add for mean aggregation: acc[dst[e]] += x[src[e]] (per feature),
// cnt[dst[e]] += 1 once per edge. dim = 1<<ld features.
// ---------------------------------------------------------------------------
__global__ void scatter_add(const float* __restrict__ x,
                            const int* __restrict__ src,
                            const int* __restrict__ dst,
                            float* __restrict__ acc, int* __restrict__ cnt,
                            long total, int ld)
{
    long idx = (long)blockIdx.x * blockDim.x + threadIdx.x;
    if (idx >= total) return;
    const int  dim = 1 << ld;
    const long e = idx >> ld;
    const int  h = (int)(idx & (dim - 1));
    const int  sI = src[e], dI = dst[e];
    atomicAdd(&acc[(size_t)dI * dim + h], x[(size_t)sI * dim + h]);
    if (h == 0 && cnt) atomicAdd(&cnt[dI], 1);
}

// u[k] = sum_h a[h] * W[h*K + k]   (tiny: one block)
__global__ void make_uvec(const float* __restrict__ W, const float* __restrict__ a,
                          float* __restrict__ u, int K)
{
    int k = threadIdx.x;
    if (k < K) {
        float s = 0.0f;
        for (int h = 0; h < Hdim; ++h) s += a[h] * W[(size_t)h * K + k];
        u[k] = s;
    }
}

// out[i] = dot(X[i,:K], v) (+ bscalar)
__global__ void rowdot(const float* __restrict__ X, const float* __restrict__ v,
                       const float* __restrict__ bscalar, float* __restrict__ out,
                       int N, int K)
{
    int i = blockIdx.x * blockDim.x + threadIdx.x;
    if (i >= N) return;
    float s = bscalar ? bscalar[0] : 0.0f;
    const float* xp = X + (size_t)i * K;
    for (int k = 0; k < K; ++k) s += xp[k] * v[k];
    out[i] = s;
}

// per-edge attention logit + segment max (encoded atomicMax)
__global__ void gat_logits(const float* __restrict__ es, const float* __restrict__ ed,
                           const float* __restrict__ eattr, const float* __restrict__ vedge,
                           const int* __restrict__ src, const int* __restrict__ dst,
                           float* __restrict__ logits, unsigned* __restrict__ mseg, int E)
{
    int e = blockIdx.x * blockDim.x + threadIdx.x;
    if (e >= E) return;
    float l = es[src[e]] + ed[dst[e]];
    const float* ep = eattr + (size_t)e * 8;
#pragma unroll
    for (int j = 0; j < 8; ++j) l += ep[j] * vedge[j];
    l = l > 0.0f ? l : 0.2f * l;               // leaky_relu(0.2)
    logits[e] = l;
    atomicMax(&mseg[dst[e]], enc_f(l));
}

// ex[e] = exp(l - m[dst]);  den[dst] += ex
__global__ void gat_exp(const float* __restrict__ logits, const unsigned* __restrict__ mseg,
                        const int* __restrict__ dst, float* __restrict__ ex,
                        float* __restrict__ den, int E)
{
    int e = blockIdx.x * blockDim.x + threadIdx.x;
    if (e >= E) return;
    const int d = dst[e];
    float x = __expf(logits[e] - dec_f(mseg[d]));
    ex[e] = x;
    atomicAdd(&den[d], x);
}

// hist[dst, h] += (ex/den) * xs[src, h]
__global__ void gat_hist(const float* __restrict__ ex, const float* __restrict__ den,
                         const float* __restrict__ xs, const int* __restrict__ src,
                         const int* __restrict__ dst, float* __restrict__ hist, long total)
{
    long idx = (long)blockIdx.x * blockDim.x + threadIdx.x;
    if (idx >= total) return;
    const long e = idx >> 6;
    const int  h = (int)(idx & 63);
    const int  d = dst[e];
    const float alpha = ex[e] / fmaxf(den[d], 1e-16f);
    atomicAdd(&hist[(size_t)d * Hdim + h], alpha * xs[(size_t)src[e] * Hdim + h]);
}

// x = relu(x + b[col])
__global__ void bias_relu(float* __restrict__ x, const float* __restrict__ b, long total)
{
    long idx = (long)blockIdx.x * blockDim.x + threadIdx.x;
    if (idx >= total) return;
    x[idx] = fmaxf(x[idx] + b[idx & (Hdim - 1)], 0.0f);
}

// ---------------------------------------------------------------------------
extern "C" void kernel_launch(void* const* d_in, const int* in_sizes, int n_in,
                              void* d_out, int out_size, void* d_ws, size_t ws_size,
                              hipStream_t stream)
{
    (void)in_sizes; (void)n_in; (void)out_size; (void)ws_size;

    const float* x_game   = (const float*)d_in[0];
    const float* x_state  = (const float*)d_in[1];
    const float* e_attr   = (const float*)d_in[2];
    const float* w_gv0_l  = (const float*)d_in[3];
    const float* w_gv0_r  = (const float*)d_in[4];
    const float* b_gv0    = (const float*)d_in[5];
    const float* w_gv1_l  = (const float*)d_in[6];
    const float* w_gv1_r  = (const float*)d_in[7];
    const float* b_gv1    = (const float*)d_in[8];
    const float* w_sv0_l  = (const float*)d_in[9];
    const float* w_sv0_r  = (const float*)d_in[10];
    const float* b_sv0    = (const float*)d_in[11];
    const float* w_sv1_l  = (const float*)d_in[12];
    const float* w_sv1_r  = (const float*)d_in[13];
    const float* b_sv1    = (const float*)d_in[14];
    const float* w_att_src= (const float*)d_in[15];
    const float* w_att_dst= (const float*)d_in[16];
    const float* w_att_edge=(const float*)d_in[17];
    const float* a_src    = (const float*)d_in[18];
    const float* a_dst    = (const float*)d_in[19];
    const float* a_edge   = (const float*)d_in[20];
    const float* b_att    = (const float*)d_in[21];
    const float* w_in_l   = (const float*)d_in[22];
    const float* w_in_r   = (const float*)d_in[23];
    const float* b_in     = (const float*)d_in[24];
    const float* w_mlp    = (const float*)d_in[25];
    const float* b_mlp    = (const float*)d_in[26];
    const int*   ei_gg    = (const int*)d_in[27];   // [2,E]: src=ei, dst=ei+E
    const int*   ei_ss    = (const int*)d_in[28];
    const int*   ei_hist  = (const int*)d_in[29];
    const int*   ei_in    = (const int*)d_in[30];
    float* out = (float*)d_out;

    // ---- workspace carve-up (all 256B aligned) ----
    char* p = (char*)d_ws;
    auto carve = [&](size_t bytes) {
        void* r = (void*)p;
        p += (bytes + 255) & ~(size_t)255;
        return r;
    };
    const size_t NF = (size_t)NG_ * Hdim;       // node feature block
    float* W1 = (float*)carve(NF * 4);
    float* W2 = (float*)carve(NF * 4);
    float* W3 = (float*)carve(NF * 4);
    float* W4 = (float*)carve(NF * 4);
    float* EB1 = (float*)carve((size_t)E_TOT * 4);   // logits
    float* EB2 = (float*)carve((size_t)E_TOT * 4);   // exp
    int*      cnt  = (int*)carve((size_t)NG_ * 4);
    unsigned* mseg = (unsigned*)carve((size_t)NS_ * 4);
    float*    den  = (float*)carve((size_t)NS_ * 4);
    float*    es   = (float*)carve((size_t)NG_ * 4);
    float*    ed   = (float*)carve((size_t)NS_ * 4);
    float*    uvec = (float*)carve(Hdim * 4);
    float*    vedge= (float*)carve(8 * 4);

    const int TB = 256;
    const long tot8  = (long)E_TOT * 8;
    const long tot64 = (long)E_TOT * 64;
    const int gTiles = (NG_ + 15) / 16;
    const int sTiles = (NS_ + 15) / 16;

    // ================= game SAGE stack =================
    hipMemsetAsync(W2, 0, (size_t)NG_ * 8 * 4, stream);
    hipMemsetAsync(cnt, 0, (size_t)NG_ * 4, stream);
    scatter_add<<<(int)((tot8 + TB - 1) / TB), TB, 0, stream>>>(
        x_game, ei_gg, ei_gg + E_TOT, W2, cnt, tot8, 3);
    sage_wmma<<<gTiles, 128, 0, stream>>>(W2, cnt, w_gv0_l, x_game, w_gv0_r,
                                          b_gv0, W1, NG_, 8, 1);       // g0 -> W1
    hipMemsetAsync(W2, 0, NF * 4, stream);
    hipMemsetAsync(cnt, 0, (size_t)NG_ * 4, stream);
    scatter_add<<<(int)((tot64 + TB - 1) / TB), TB, 0, stream>>>(
        W1, ei_gg, ei_gg + E_TOT, W2, cnt, tot64, 6);
    sage_wmma<<<gTiles, 128, 0, stream>>>(W2, cnt, w_gv1_l, W1, w_gv1_r,
                                          b_gv1, W2, NG_, 64, 1);      // g -> W2

    // ================= state SAGE stack =================
    hipMemsetAsync(W1, 0, (size_t)NS_ * 8 * 4, stream);
    hipMemsetAsync(cnt, 0, (size_t)NS_ * 4, stream);
    scatter_add<<<(int)((tot8 + TB - 1) / TB), TB, 0, stream>>>(
        x_state, ei_ss, ei_ss + E_TOT, W1, cnt, tot8, 3);
    sage_wmma<<<sTiles, 128, 0, stream>>>(W1, cnt, w_sv0_l, x_state, w_sv0_r,
                                          b_sv0, W3, NS_, 8, 1);       // s0 -> W3
    hipMemsetAsync(W1, 0, NF * 4, stream);
    hipMemsetAsync(cnt, 0, (size_t)NS_ * 4, stream);
    scatter_add<<<(int)((tot64 + TB - 1) / TB), TB, 0, stream>>>(
        W3, ei_ss, ei_ss + E_TOT, W1, cnt, tot64, 6);
    sage_wmma<<<sTiles, 128, 0, stream>>>(W1, cnt, w_sv1_l, W3, w_sv1_r,
                                          b_sv1, W1, NS_, 64, 1);      // s -> W1

    // ================= GAT game -> state =================
    // xs = g @ W_src^T  (plain GEMM, no bias/relu) -> W3
    sage_wmma<<<gTiles, 128, 0, stream>>>(W2, nullptr, w_att_src, nullptr,
                                          nullptr, nullptr, W3, NG_, 64, 0);
    make_uvec<<<1, 64, 0, stream>>>(w_att_dst, a_dst, uvec, 64);   // W_dst^T a_dst
    make_uvec<<<1, 64, 0, stream>>>(w_att_edge, a_edge, vedge, 8); // W_edge^T a_edge
    rowdot<<<(NG_ + TB - 1) / TB, TB, 0, stream>>>(W3, a_src, nullptr, es, NG_, 64);
    rowdot<<<(NS_ + TB - 1) / TB, TB, 0, stream>>>(W1, uvec, nullptr, ed, NS_, 64);

    hipMemsetAsync(mseg, 0, (size_t)NS_ * 4, stream);   // 0 < enc(any finite float)
    hipMemsetAsync(den, 0, (size_t)NS_ * 4, stream);
    gat_logits<<<(E_TOT + TB - 1) / TB, TB, 0, stream>>>(
        es, ed, e_attr, vedge, ei_hist, ei_hist + E_TOT, EB1, mseg, E_TOT);
    gat_exp<<<(E_TOT + TB - 1) / TB, TB, 0, stream>>>(
        EB1, mseg, ei_hist + E_TOT, EB2, den, E_TOT);
    hipMemsetAsync(W1, 0, NF * 4, stream);              // s dead; reuse for hist
    gat_hist<<<(int)((tot64 + TB - 1) / TB), TB, 0, stream>>>(
        EB2, den, W3, ei_hist, ei_hist + E_TOT, W1, tot64);
    bias_relu<<<(int)(((size_t)NS_ * 64 + TB - 1) / TB), TB, 0, stream>>>(
        W1, b_att, (long)NS_ * 64);                     // hist -> W1

    // ================= bipartite in1 SAGE =================
    hipMemsetAsync(W4, 0, NF * 4, stream);
    hipMemsetAsync(cnt, 0, (size_t)NS_ * 4, stream);
    scatter_add<<<(int)((tot64 + TB - 1) / TB), TB, 0, stream>>>(
        W2, ei_in, ei_in + E_TOT, W4, cnt, tot64, 6);
    sage_wmma<<<sTiles, 128, 0, stream>>>(W4, cnt, w_in_l, W1, w_in_r,
                                          b_in, W4, NS_, 64, 1);       // in_x -> W4

    // ================= MLP head =================
    rowdot<<<(NS_ + TB - 1) / TB, TB, 0, stream>>>(W4, w_mlp, b_mlp, out, NS_, 64);
}